// MultiboxLoss_37666863186566
// MI455X (gfx1250) — compile-verified
//
#include <hip/hip_runtime.h>
#include <hip/hip_bf16.h>
#include <math.h>

typedef __attribute__((ext_vector_type(2))) float v2f;
typedef __attribute__((ext_vector_type(8))) float v8f;

#define P_PRIORS 8732
#define N_CLASSES 21
#define N_OBJ 16
#define THRESH 0.5f
#define BLOCK 256
#define NWAVES (BLOCK / 32)

// Sum 64 f32 values (2 per lane across a wave32) using V_WMMA_F32_16X16X4_F32.
// A(16x4): per the ISA layout, VGPR0/1 of each lane hold 2 distinct A elements
// (M = lane%16, K = vgpr + 2*(lane/16)).  B = ones(4x16) -> D rows are row-sums.
// Each lane sums its 8 C/D VGPRs (covering rows 0-7 or 8-15 depending on half),
// then xor-shuffle across the lane-16 boundary gives the full sum in every lane.
__device__ __forceinline__ float wave_sum_wmma(float x, float y) {
    v2f a; a.x = x;    a.y = y;
    v2f b; b.x = 1.0f; b.y = 1.0f;
    v8f c = {};
    c = __builtin_amdgcn_wmma_f32_16x16x4_f32(false, a, false, b, (short)0, c,
                                              false, false);
    float s = c[0] + c[1] + c[2] + c[3] + c[4] + c[5] + c[6] + c[7];
    s += __shfl_xor(s, 16, 32);
    return s;
}

// Block-wide sum; must be called by ALL threads (WMMA requires EXEC all-ones).
__device__ __forceinline__ float block_sum(float x, float y, float* s_red,
                                           int nwaves) {
    float s = wave_sum_wmma(x, y);
    const int lane = threadIdx.x & 31;
    const int wid  = threadIdx.x >> 5;
    if (lane == 0) s_red[wid] = s;
    __syncthreads();
    float tot = 0.f;
    for (int i = 0; i < nwaves; ++i) tot += s_red[i];
    __syncthreads();
    return tot;
}

__global__ __launch_bounds__(BLOCK) void multibox_match_kernel(
    const float* __restrict__ locs,    // (B,P,4)
    const float* __restrict__ scores,  // (B,P,21)
    const float* __restrict__ boxes,   // (B,16,4) xyxy
    const int*   __restrict__ labels,  // (B,16)
    const float* __restrict__ priors,  // (P,4) cxcywh
    float* __restrict__ partial)       // (B,4): n_pos, loc_sum, conf_pos, conf_hard
{
    __shared__ float               s_val[P_PRIORS];   // overlap, then ce_neg
    __shared__ unsigned char       s_obj[P_PRIORS];
    __shared__ unsigned long long  s_best[N_OBJ];
    __shared__ float               s_box[N_OBJ * 4];
    __shared__ int                 s_lab[N_OBJ];
    __shared__ float               s_red[NWAVES];

    const int b   = blockIdx.x;
    const int tid = threadIdx.x;

    if (tid < N_OBJ * 4) s_box[tid] = boxes[b * N_OBJ * 4 + tid];
    if (tid < N_OBJ) { s_lab[tid] = labels[b * N_OBJ + tid]; s_best[tid] = 0ull; }
    __syncthreads();

    // ---- Phase A: Jaccard matching ------------------------------------------
    unsigned long long best[N_OBJ];
#pragma unroll
    for (int o = 0; o < N_OBJ; ++o) best[o] = 0ull;

    for (int p = tid; p < P_PRIORS; p += BLOCK) {
        float4 pr  = ((const float4*)priors)[p];
        float px0 = pr.x - 0.5f * pr.z, py0 = pr.y - 0.5f * pr.w;
        float px1 = pr.x + 0.5f * pr.z, py1 = pr.y + 0.5f * pr.w;
        float ap  = pr.z * pr.w;
        float bi = -1.0f; int bo = 0;
#pragma unroll
        for (int o = 0; o < N_OBJ; ++o) {
            float x0 = s_box[o * 4 + 0], y0 = s_box[o * 4 + 1];
            float x1 = s_box[o * 4 + 2], y1 = s_box[o * 4 + 3];
            float iw    = fminf(x1, px1) - fmaxf(x0, px0);
            float ih    = fminf(y1, py1) - fmaxf(y0, py0);
            float inter = fmaxf(iw, 0.f) * fmaxf(ih, 0.f);
            float ab    = (x1 - x0) * (y1 - y0);
            float iou   = inter / (ab + ap - inter);
            if (iou > bi) { bi = iou; bo = o; }           // first-max tiebreak
            unsigned long long pk =
                ((unsigned long long)__float_as_uint(iou) << 32) |
                (unsigned long long)(0xFFFFFFFFu - (unsigned)p);
            if (pk > best[o]) best[o] = pk;               // smallest p wins ties
        }
        s_val[p] = bi;
        s_obj[p] = (unsigned char)bo;
    }
#pragma unroll
    for (int o = 0; o < N_OBJ; ++o) atomicMax(&s_best[o], best[o]);
    __syncthreads();

    // Forced matches (sequential: duplicate priors -> last object wins)
    if (tid == 0) {
        for (int o = 0; o < N_OBJ; ++o) {
            unsigned pstar = 0xFFFFFFFFu - (unsigned)(s_best[o] & 0xFFFFFFFFull);
            s_obj[pstar] = (unsigned char)o;
            s_val[pstar] = 1.0f;
        }
    }
    __syncthreads();

    // ---- Phase B: per-prior CE + loc loss -----------------------------------
    float npos_f = 0.f, loc_acc = 0.f, cpos = 0.f;
    for (int p = tid; p < P_PRIORS; p += BLOCK) {
        const int   o  = s_obj[p];
        const float ov = s_val[p];
        const int lab  = (ov < THRESH) ? 0 : s_lab[o];

        const float* sc = scores + ((size_t)b * P_PRIORS + p) * N_CLASSES;
        float m = sc[0];
#pragma unroll
        for (int j = 1; j < N_CLASSES; ++j) m = fmaxf(m, sc[j]);
        float se = 0.f;
#pragma unroll
        for (int j = 0; j < N_CLASSES; ++j) se += expf(sc[j] - m);
        float ce = m + logf(se) - sc[lab];

        if (lab != 0) {
            npos_f += 1.f;
            cpos   += ce;
            float4 pl = ((const float4*)locs)[(size_t)b * P_PRIORS + p];
            float4 pr = ((const float4*)priors)[p];
            float x0 = s_box[o * 4 + 0], y0 = s_box[o * 4 + 1];
            float x1 = s_box[o * 4 + 2], y1 = s_box[o * 4 + 3];
            float t0 = 10.f * (0.5f * (x0 + x1) - pr.x) / pr.z;
            float t1 = 10.f * (0.5f * (y0 + y1) - pr.y) / pr.w;
            float t2 = 5.f * logf((x1 - x0) / pr.z);
            float t3 = 5.f * logf((y1 - y0) / pr.w);
            float d, ad;
            d = pl.x - t0; ad = fabsf(d); loc_acc += (ad < 1.f) ? 0.5f * d * d : ad - 0.5f;
            d = pl.y - t1; ad = fabsf(d); loc_acc += (ad < 1.f) ? 0.5f * d * d : ad - 0.5f;
            d = pl.z - t2; ad = fabsf(d); loc_acc += (ad < 1.f) ? 0.5f * d * d : ad - 0.5f;
            d = pl.w - t3; ad = fabsf(d); loc_acc += (ad < 1.f) ? 0.5f * d * d : ad - 0.5f;
            s_val[p] = 0.f;        // positives excluded from hard negatives
        } else {
            s_val[p] = ce;         // ce_neg
        }
    }
    __syncthreads();

    const float np     = block_sum(npos_f,  0.f, s_red, NWAVES);
    const float locsum = block_sum(loc_acc, 0.f, s_red, NWAVES);
    const float cps    = block_sum(cpos,    0.f, s_red, NWAVES);

    // ---- Phase C: exact top-K sum via bitwise threshold search --------------
    const int K = 3 * (int)(np + 0.5f);
    unsigned t = 0u;
    for (int bit = 31; bit >= 0; --bit) {
        const unsigned cand = t | (1u << bit);
        float cnt = 0.f;
        for (int p = tid; p < P_PRIORS; p += BLOCK)
            if (__float_as_uint(s_val[p]) >= cand) cnt += 1.f;
        const float total = block_sum(cnt, 0.f, s_red, NWAVES);
        if (total >= (float)K) t = cand;   // monotone predicate; all lanes agree
    }
    const float v = __uint_as_float(t);    // K-th largest (or 0 if K > #nonzero)
    float sgt = 0.f, cgt = 0.f;
    for (int p = tid; p < P_PRIORS; p += BLOCK) {
        const float x = s_val[p];
        if (__float_as_uint(x) > t) { sgt += x; cgt += 1.f; }
    }
    const float sum_gt = block_sum(sgt, 0.f, s_red, NWAVES);
    const float cnt_gt = block_sum(cgt, 0.f, s_red, NWAVES);
    const float chard  = (K > 0) ? (sum_gt + ((float)K - cnt_gt) * v) : 0.f;

    if (tid == 0) {
        partial[b * 4 + 0] = np;
        partial[b * 4 + 1] = locsum;
        partial[b * 4 + 2] = cps;
        partial[b * 4 + 3] = chard;
    }
}

__global__ __launch_bounds__(128) void multibox_reduce_kernel(
    const float* __restrict__ partial, float* __restrict__ out, int B)
{
    __shared__ float s_red[4];
    const int tid = threadIdx.x;
    float np = 0.f, lc = 0.f, cp = 0.f, ch = 0.f;
    for (int i = tid; i < B; i += 128) {
        np += partial[i * 4 + 0];
        lc += partial[i * 4 + 1];
        cp += partial[i * 4 + 2];
        ch += partial[i * 4 + 3];
    }
    const float tp = block_sum(np, 0.f, s_red, 4);
    const float tl = block_sum(lc, 0.f, s_red, 4);
    const float tc = block_sum(cp, 0.f, s_red, 4);
    const float th = block_sum(ch, 0.f, s_red, 4);
    if (tid == 0) out[0] = (tc + th) / tp + tl / (4.f * tp);
}

extern "C" void kernel_launch(void* const* d_in, const int* in_sizes, int n_in,
                              void* d_out, int out_size, void* d_ws, size_t ws_size,
                              hipStream_t stream) {
    const float* locs   = (const float*)d_in[0];
    const float* scores = (const float*)d_in[1];
    const float* boxes  = (const float*)d_in[2];
    const int*   labels = (const int*)d_in[3];
    const float* priors = (const float*)d_in[4];
    const int B = in_sizes[0] / (P_PRIORS * 4);

    float* partial = (float*)d_ws;   // B*4 floats

    multibox_match_kernel<<<B, BLOCK, 0, stream>>>(locs, scores, boxes, labels,
                                                   priors, partial);
    multibox_reduce_kernel<<<1, 128, 0, stream>>>(partial, (float*)d_out, B);
}